// HybridQLSTMQuantum_65481071396138
// MI455X (gfx1250) — compile-verified
//
#include <hip/hip_runtime.h>
#include <hip/hip_bf16.h>

typedef __attribute__((ext_vector_type(2))) float v2f;
typedef __attribute__((ext_vector_type(8))) float v8f;

#define SEQ 512
#define BATCH 256
#define DIN 256
#define NQ 4
#define DTOT 260           // DIN + NQ
#define MROWS (SEQ * BATCH)   // 131072
#define NCOLS 16              // 4 gates x 4 wires

// ---------------------------------------------------------------------------
// Phase 1: preA[m][n] = sum_k x[m][k] * Wpack[k][n] + bias[n]
//   m = t*BATCH + b (131072 rows), n = gate*4 + wire, k over DIN=256.
//   Full-fp32 WMMA (V_WMMA_F32_16X16X4_F32), one 16x16 tile per wave,
//   64 WMMA k-steps. Memory bound: streams x (128 MB) once.
// ---------------------------------------------------------------------------
__global__ __launch_bounds__(256) void qlstm_angles_wmma(
    const float* __restrict__ x,
    const float* __restrict__ Wf, const float* __restrict__ bf,
    const float* __restrict__ Wi, const float* __restrict__ bi,
    const float* __restrict__ Wg, const float* __restrict__ bg,
    const float* __restrict__ Wo, const float* __restrict__ bo,
    float* __restrict__ preA)
{
    __shared__ float Wl[DIN * NCOLS];   // packed [k][n], 16 KB
    __shared__ float Bl[NCOLS];

    const int tid = threadIdx.x;

    // Stage packed weights into LDS: Wl[k*16 + n] = W_gate[wire*260 + k]
    for (int idx = tid; idx < DIN * NCOLS; idx += 256) {
        const int n    = idx & 15;
        const int k    = idx >> 4;
        const int gate = n >> 2;
        const int wire = n & 3;
        const float* wp = (gate == 0) ? Wf : (gate == 1) ? Wi : (gate == 2) ? Wg : Wo;
        Wl[idx] = wp[wire * DTOT + k];
    }
    if (tid < NCOLS) {
        const int gate = tid >> 2;
        const float* bp = (gate == 0) ? bf : (gate == 1) ? bi : (gate == 2) ? bg : bo;
        Bl[tid] = bp[tid & 3];
    }
    __syncthreads();

    const int lane  = tid & 31;
    const int wave  = tid >> 5;
    const int mtile = blockIdx.x * 8 + wave;       // 1024 blocks * 8 waves = 8192 tiles
    const int m0    = mtile * 16;
    const int n     = lane & 15;                   // N for B/C/D frags, M-row for A frag
    const int h     = lane >> 4;                   // lane half selects K sub-pair

    const float* __restrict__ xrow = x + (size_t)(m0 + n) * DIN + 2 * h;

    v8f acc = {};
    #pragma unroll 4
    for (int k0 = 0; k0 < DIN; k0 += 4) {
        // A frag (16x4 f32): lanes 0-15 hold K=k0,k0+1 of row M=lane;
        //                    lanes 16-31 hold K=k0+2,k0+3 of row M=lane-16.
        v2f a, b;
        a[0] = xrow[k0];
        a[1] = xrow[k0 + 1];
        // B frag (4x16 f32): VGPR0 = row K=k0 (+2 for upper half), VGPR1 = next K.
        b[0] = Wl[(k0 + 2 * h) * NCOLS + n];
        b[1] = Wl[(k0 + 2 * h + 1) * NCOLS + n];
        acc = __builtin_amdgcn_wmma_f32_16x16x4_f32(
            /*neg_a=*/false, a, /*neg_b=*/false, b,
            /*c_mod=*/(short)0, acc, /*reuse_a=*/false, /*reuse_b=*/false);
    }

    // D layout: VGPR v -> M = v + 8*h, N = lane&15
    const float bias = Bl[n];
    float* __restrict__ outp = preA + ((size_t)m0 + 8 * h) * NCOLS + n;
    #pragma unroll
    for (int v = 0; v < 8; ++v) {
        outp[(size_t)v * NCOLS] = acc[v] + bias;
    }
}

// ---------------------------------------------------------------------------
// Phase 2: sequential scan over 512 timesteps. One lane per (sample, gate);
// 8 samples per wave32, 32 blocks (samples are independent -> no global sync).
// Quantum circuit evaluated in closed form:
//   m0 = cos(a0+th0), m1 = cos(th1)cos(a1), m2 = cos(th2), m3 = cos(a3+th3)
//   z  = (m1 m2 m3, m0 m1, m0 m1 m2, m0 m1 m2 m3)
// Cross-gate exchange for the LSTM cell update via wave shuffles.
// ---------------------------------------------------------------------------
__global__ __launch_bounds__(32) void qlstm_scan(
    const float* __restrict__ preA,
    const float* __restrict__ Wf, const float* __restrict__ thf,
    const float* __restrict__ Wi, const float* __restrict__ thi,
    const float* __restrict__ Wg, const float* __restrict__ thg,
    const float* __restrict__ Wo, const float* __restrict__ tho,
    float* __restrict__ out)
{
    const int lane   = threadIdx.x;
    const int gate   = lane & 3;
    const int sample = blockIdx.x * 8 + (lane >> 2);

    const float* W  = (gate == 0) ? Wf  : (gate == 1) ? Wi  : (gate == 2) ? Wg  : Wo;
    const float* th = (gate == 0) ? thf : (gate == 1) ? thi : (gate == 2) ? thg : tho;

    // Recurrent weights: Wh[w][j] = W[w*260 + 256 + j]
    float Wh[NQ][NQ];
    #pragma unroll
    for (int w = 0; w < NQ; ++w)
        #pragma unroll
        for (int j = 0; j < NQ; ++j)
            Wh[w][j] = W[w * DTOT + DIN + j];

    const float th0 = th[0];
    const float ct1 = __cosf(th[1]) != __cosf(th[1]) ? 0.f : cosf(th[1]);
    const float ct2 = cosf(th[2]);
    const float th3 = th[3];

    float hx[NQ] = {0.f, 0.f, 0.f, 0.f};
    float cx[NQ] = {0.f, 0.f, 0.f, 0.f};

    const float4* __restrict__ pa = (const float4*)preA;
    const int base = lane & ~3;

    for (int t = 0; t < SEQ; ++t) {
        const float4 p = pa[(size_t)(t * BATCH + sample) * NQ + gate];

        const float a0 = p.x + Wh[0][0]*hx[0] + Wh[0][1]*hx[1] + Wh[0][2]*hx[2] + Wh[0][3]*hx[3];
        const float a1 = p.y + Wh[1][0]*hx[0] + Wh[1][1]*hx[1] + Wh[1][2]*hx[2] + Wh[1][3]*hx[3];
        // a2 (wire 2, RZ encoder) provably does not affect any Z expectation.
        const float a3 = p.w + Wh[3][0]*hx[0] + Wh[3][1]*hx[1] + Wh[3][2]*hx[2] + Wh[3][3]*hx[3];

        const float m0 = cosf(a0 + th0);
        const float m1 = ct1 * cosf(a1);
        const float m2 = ct2;
        const float m3 = cosf(a3 + th3);

        float z[NQ];
        z[0] = m1 * m2 * m3;
        z[1] = m0 * m1;
        z[2] = z[1] * m2;
        z[3] = z[2] * m3;

        float act[NQ];
        #pragma unroll
        for (int w = 0; w < NQ; ++w) {
            act[w] = (gate == 2) ? tanhf(z[w]) : 1.0f / (1.0f + __expf(-z[w]));
        }

        // Exchange gate values across the 4 lanes of this sample.
        #pragma unroll
        for (int w = 0; w < NQ; ++w) {
            const float fv = __shfl(act[w], base + 0, 32);
            const float iv = __shfl(act[w], base + 1, 32);
            const float gv = __shfl(act[w], base + 2, 32);
            const float ov = __shfl(act[w], base + 3, 32);
            cx[w] = fv * cx[w] + iv * gv;
            hx[w] = ov * tanhf(cx[w]);
        }

        if (gate == 0) {
            float4 o4;
            o4.x = hx[0]; o4.y = hx[1]; o4.z = hx[2]; o4.w = hx[3];
            ((float4*)out)[(size_t)t * BATCH + sample] = o4;
        }
    }

    if (gate == 0) {
        float4 h4, c4;
        h4.x = hx[0]; h4.y = hx[1]; h4.z = hx[2]; h4.w = hx[3];
        c4.x = cx[0]; c4.y = cx[1]; c4.z = cx[2]; c4.w = cx[3];
        ((float4*)(out + (size_t)SEQ * BATCH * NQ))[sample] = h4;
        ((float4*)(out + (size_t)SEQ * BATCH * NQ + BATCH * NQ))[sample] = c4;
    }
}

extern "C" void kernel_launch(void* const* d_in, const int* in_sizes, int n_in,
                              void* d_out, int out_size, void* d_ws, size_t ws_size,
                              hipStream_t stream) {
    // setup_inputs order: x, Wf, bf, thf, Wi, bi, thi, Wg, bg, thg, Wo, bo, tho
    const float* x   = (const float*)d_in[0];
    const float* Wf  = (const float*)d_in[1];
    const float* bf  = (const float*)d_in[2];
    const float* thf = (const float*)d_in[3];
    const float* Wi  = (const float*)d_in[4];
    const float* bi  = (const float*)d_in[5];
    const float* thi = (const float*)d_in[6];
    const float* Wg  = (const float*)d_in[7];
    const float* bg  = (const float*)d_in[8];
    const float* thg = (const float*)d_in[9];
    const float* Wo  = (const float*)d_in[10];
    const float* bo  = (const float*)d_in[11];
    const float* tho = (const float*)d_in[12];

    float* preA = (float*)d_ws;   // MROWS * 16 floats = 8 MB scratch
    float* out  = (float*)d_out;

    // 8192 M-tiles, 8 waves (tiles) per 256-thread block -> 1024 blocks
    qlstm_angles_wmma<<<dim3(MROWS / (16 * 8)), dim3(256), 0, stream>>>(
        x, Wf, bf, Wi, bi, Wg, bg, Wo, bo, preA);

    // 256 samples * 4 gate-lanes = 1024 lanes -> 32 wave32 blocks
    qlstm_scan<<<dim3(BATCH / 8), dim3(32), 0, stream>>>(
        preA, Wf, thf, Wi, thi, Wg, thg, Wo, tho, out);
}